// VSIE_23175643529624
// MI455X (gfx1250) — compile-verified
//
#include <hip/hip_runtime.h>
#include <math.h>

typedef __attribute__((ext_vector_type(2))) float v2f;
typedef __attribute__((ext_vector_type(8))) float v8f;

#define Lrows 8192   // T*N
#define IMGINV 0.125f

__device__ __forceinline__ v8f wmma4(v2f a, v2f b, v8f c) {
  // D = A(16x4,f32) * B(4x16,f32) + C(16x16,f32)
  return __builtin_amdgcn_wmma_f32_16x16x4_f32(
      /*neg_a=*/false, a, /*neg_b=*/false, b,
      /*c_mod=*/(short)0, c, /*reuse_a=*/false, /*reuse_b=*/false);
}

// ---------------- Stage 1: positional encoding -> xi [L,2] flat ----------------
__global__ void k_posenc(const float* __restrict__ x, float* __restrict__ xi) {
  int f = blockIdx.x * blockDim.x + threadIdx.x;   // 0..16383
  int rem = f & 8191;
  int t = rem >> 7;
  int n = rem & 127;
  // ang = t / 10000^(2n/128) = t * exp(-n * ln(10000)/64)
  float scale = __expf(-(float)n * (9.2103403719761836f / 64.0f));
  float ang = (float)t * scale;
  float pe = ((n & 1) == 0) ? sinf(ang) : cosf(ang);
  xi[f] = x[f] + pe;
}

// ---------------- Stage 2: serial LSTM (1 wave) -> Xl [L,4] ----------------
__global__ void k_lstm(const float* __restrict__ xi,
                       const float* __restrict__ W_ih, const float* __restrict__ W_hh,
                       const float* __restrict__ b_ih, const float* __restrict__ b_hh,
                       float* __restrict__ Xl) {
  int l = threadIdx.x;        // 0..31
  int g = l & 15;             // gate index for lanes 0..15 (pytorch order i,f,g,o)
  float wi0 = 0.f, wi1 = 0.f, wh0 = 0.f, wh1 = 0.f, wh2 = 0.f, wh3 = 0.f, bs = 0.f;
  if (l < 16) {
    wi0 = W_ih[g * 2 + 0]; wi1 = W_ih[g * 2 + 1];
    wh0 = W_hh[g * 4 + 0]; wh1 = W_hh[g * 4 + 1];
    wh2 = W_hh[g * 4 + 2]; wh3 = W_hh[g * 4 + 3];
    bs  = b_ih[g] + b_hh[g];
  }
  float h = 0.f, c = 0.f;     // lanes 0..3 hold h_k, c_k
  for (int t = 0; t < Lrows; ++t) {
    v2f xv = *(const v2f*)(xi + t * 2);
    float h0 = __shfl(h, 0, 32), h1 = __shfl(h, 1, 32);
    float h2 = __shfl(h, 2, 32), h3 = __shfl(h, 3, 32);
    float gate = bs + wi0 * xv.x + wi1 * xv.y
                    + wh0 * h0 + wh1 * h1 + wh2 * h2 + wh3 * h3;
    int k = l & 3;
    float gi = __shfl(gate, k,      32);
    float gf = __shfl(gate, k + 4,  32);
    float gc = __shfl(gate, k + 8,  32);
    float go = __shfl(gate, k + 12, 32);
    if (l < 4) {
      float si = 1.f / (1.f + __expf(-gi));
      float sf = 1.f / (1.f + __expf(-gf));
      float so = 1.f / (1.f + __expf(-go));
      c = sf * c + si * tanhf(gc);
      h = so * tanhf(c);
      Xl[t * 4 + l] = h;
    }
  }
}

// ------- Stage 3: compressor GEMM  cm[hw][32] = (W_comp[32,2048] @ meta)^T + b -------
// One wave per (M=32, N=16) tile; 512 x v_wmma_f32_16x16x4_f32 per accumulator.
__global__ void k_comp(const float* __restrict__ meta,  // [2048][4096]
                       const float* __restrict__ Wc,    // [32][2048]
                       const float* __restrict__ bc,    // [32]
                       float* __restrict__ cm) {        // [4096][32] channel-last
  int l = threadIdx.x;
  int n = blockIdx.x * 16 + (l & 15);  // spatial column
  int cc = 2 * (l >> 4);               // feature-pair select within K-slice
  v8f acc0 = {}; v8f acc1 = {};
  const float* wrow0 = Wc + (l & 15) * 2048 + cc;        // A rows 0..15
  const float* wrow1 = wrow0 + 16 * 2048;                // A rows 16..31
  const float* bcol  = meta + n;
  for (int kk = 0; kk < 2048; kk += 4) {
    v2f a0 = *(const v2f*)(wrow0 + kk);
    v2f a1 = *(const v2f*)(wrow1 + kk);
    v2f b  = { bcol[(size_t)(kk + cc) * 4096], bcol[(size_t)(kk + cc + 1) * 4096] };
    acc0 = wmma4(a0, b, acc0);
    acc1 = wmma4(a1, b, acc1);
  }
  int rbase = 8 * (l >> 4);
  float* dst = cm + n * 32;
  #pragma unroll
  for (int r = 0; r < 8; ++r) {
    int row0 = r + rbase;
    int row1 = 16 + r + rbase;
    dst[row0] = acc0[r] + bc[row0];
    dst[row1] = acc1[r] + bc[row1];
  }
}

// ------- Stage 4: grid_sample + value_fusion + fc/fc2/fc3 -> Q,K,V [L,8] -------
__global__ void k_qkv(const float* __restrict__ x, const float* __restrict__ xi,
                      const float* __restrict__ Xl, const float* __restrict__ cm,
                      const float* __restrict__ Wvf, const float* __restrict__ bvf,
                      const float* __restrict__ Wfc, const float* __restrict__ bfc,
                      const float* __restrict__ Wfc2, const float* __restrict__ bfc2,
                      const float* __restrict__ Wfc3, const float* __restrict__ bfc3,
                      float* __restrict__ Q, float* __restrict__ K, float* __restrict__ V) {
  int l = blockIdx.x * blockDim.x + threadIdx.x;  // 0..8191  (l = t*128 + n)
  float gx = x[l] * IMGINV - 0.5f;                // channel 0 -> x coord
  float gy = x[8192 + l] * IMGINV - 0.5f;         // channel 1 -> y coord
  float x0 = floorf(gx), y0 = floorf(gy);
  float lc[32];
  #pragma unroll
  for (int c = 0; c < 32; ++c) lc[c] = 0.f;
  float tx[2] = { x0, x0 + 1.f }, ty[2] = { y0, y0 + 1.f };
  float wx[2] = { x0 + 1.f - gx, gx - x0 }, wy[2] = { y0 + 1.f - gy, gy - y0 };
  #pragma unroll
  for (int jy = 0; jy < 2; ++jy) {
    #pragma unroll
    for (int jx = 0; jx < 2; ++jx) {
      float xf = tx[jx], yf = ty[jy];
      bool valid = (xf >= 0.f) & (xf <= 63.f) & (yf >= 0.f) & (yf <= 63.f);
      float w = wx[jx] * wy[jy] * (valid ? 1.f : 0.f);
      int xc = (int)fminf(fmaxf(xf, 0.f), 63.f);
      int yc = (int)fminf(fmaxf(yf, 0.f), 63.f);
      const float* p = cm + (yc * 64 + xc) * 32;
      #pragma unroll
      for (int c = 0; c < 32; ++c) lc[c] += w * p[c];
    }
  }
  float h0 = Xl[l * 4 + 0], h1 = Xl[l * 4 + 1], h2 = Xl[l * 4 + 2], h3 = Xl[l * 4 + 3];
  float X2[4];
  #pragma unroll
  for (int o = 0; o < 4; ++o) {
    const float* wr = Wvf + o * 36;
    float s = bvf[o] + wr[0] * h0 + wr[1] * h1 + wr[2] * h2 + wr[3] * h3;
    #pragma unroll
    for (int c = 0; c < 32; ++c) s += wr[4 + c] * lc[c];
    X2[o] = s;
  }
  float xi0 = xi[l * 2], xi1 = xi[l * 2 + 1];
  #pragma unroll
  for (int d = 0; d < 8; ++d) {
    Q[l * 8 + d] = bfc[d] + Wfc[d * 4] * X2[0] + Wfc[d * 4 + 1] * X2[1]
                          + Wfc[d * 4 + 2] * X2[2] + Wfc[d * 4 + 3] * X2[3];
    K[l * 8 + d] = bfc2[d] + Wfc2[d * 2] * xi0 + Wfc2[d * 2 + 1] * xi1;
    V[l * 8 + d] = bfc3[d] + Wfc3[d * 2] * xi0 + Wfc3[d * 2 + 1] * xi1;
  }
}

// ------- Stage 5: fused attention  out = thr(sigmoid(QK^T) @ V) @ W_out^T + b -------
// One wave owns 16 output rows (i-tile). Score tile computed TRANSPOSED
// (T = K_tile @ Q_tile^T) so each lane owns column i and accumulates its own
// output row locally; single shfl merges the two j-halves at the end.
__global__ void k_attn(const float* __restrict__ Q, const float* __restrict__ K,
                       const float* __restrict__ Vv, const float* __restrict__ Wout,
                       const float* __restrict__ bout, float* __restrict__ out) {
  int l = threadIdx.x;
  int i0 = blockIdx.x * 16;
  int m = l & 15;
  int cc = 2 * (l >> 4);
  const float* qrow = Q + (i0 + m) * 8 + cc;     // B operand: col i = lane%16
  v2f bq0 = { qrow[0], qrow[1] };                // k-slice 0..3
  v2f bq1 = { qrow[4], qrow[5] };                // k-slice 4..7
  float acc[8];
  #pragma unroll
  for (int d = 0; d < 8; ++d) acc[d] = 0.f;
  int jh8 = 8 * (l >> 4);
  for (int j0 = 0; j0 < Lrows; j0 += 16) {
    const float* krow = K + (j0 + m) * 8 + cc;   // A operand: row j = lane%16
    v2f a0 = { krow[0], krow[1] };
    v2f a1 = { krow[4], krow[5] };
    v8f z = {};
    v8f t = wmma4(a0, bq0, z);
    t = wmma4(a1, bq1, t);                       // T[j][i] = K.Q dot, d=8
    #pragma unroll
    for (int r = 0; r < 8; ++r) {
      float s = 1.f / (1.f + __expf(-t[r]));     // sigmoid(score)
      const float* vr = Vv + (size_t)(j0 + r + jh8) * 8;
      #pragma unroll
      for (int d = 0; d < 8; ++d) acc[d] += s * vr[d];
    }
  }
  float full[8];
  #pragma unroll
  for (int d = 0; d < 8; ++d) full[d] = acc[d] + __shfl_down(acc[d], 16, 32);
  if (l < 16) {
    int i = i0 + l;
    float o0 = bout[0], o1 = bout[1];
    #pragma unroll
    for (int d = 0; d < 8; ++d) {
      float fd = (full[d] > 0.5f) ? full[d] : 0.f;   // threshold relu
      o0 += Wout[d] * fd;
      o1 += Wout[8 + d] * fd;
    }
    out[i * 2 + 0] = o0;
    out[i * 2 + 1] = o1;
  }
}

extern "C" void kernel_launch(void* const* d_in, const int* in_sizes, int n_in,
                              void* d_out, int out_size, void* d_ws, size_t ws_size,
                              hipStream_t stream) {
  const float* x     = (const float*)d_in[0];
  const float* meta  = (const float*)d_in[1];
  const float* W_ih  = (const float*)d_in[2];
  const float* W_hh  = (const float*)d_in[3];
  const float* b_ih  = (const float*)d_in[4];
  const float* b_hh  = (const float*)d_in[5];
  const float* W_fc  = (const float*)d_in[6];
  const float* b_fc  = (const float*)d_in[7];
  const float* W_fc2 = (const float*)d_in[8];
  const float* b_fc2 = (const float*)d_in[9];
  const float* W_fc3 = (const float*)d_in[10];
  const float* b_fc3 = (const float*)d_in[11];
  const float* W_out = (const float*)d_in[12];
  const float* b_out = (const float*)d_in[13];
  const float* W_cmp = (const float*)d_in[14];
  const float* b_cmp = (const float*)d_in[15];
  const float* W_vf  = (const float*)d_in[16];
  const float* b_vf  = (const float*)d_in[17];

  float* ws = (float*)d_ws;
  float* xi = ws;                  // 16384  floats
  float* Xl = ws + 16384;          // 32768  floats
  float* cm = ws + 49152;          // 131072 floats  [4096][32] channel-last
  float* Qb = ws + 180224;         // 65536  floats
  float* Kb = ws + 245760;         // 65536  floats
  float* Vb = ws + 311296;         // 65536  floats

  k_posenc<<<64, 256, 0, stream>>>(x, xi);
  k_lstm<<<1, 32, 0, stream>>>(xi, W_ih, W_hh, b_ih, b_hh, Xl);
  k_comp<<<256, 32, 0, stream>>>(meta, W_cmp, b_cmp, cm);
  k_qkv<<<32, 256, 0, stream>>>(x, xi, Xl, cm, W_vf, b_vf, W_fc, b_fc,
                                W_fc2, b_fc2, W_fc3, b_fc3, Qb, Kb, Vb);
  k_attn<<<512, 32, 0, stream>>>(Qb, Kb, Vb, W_out, b_out, (float*)d_out);
}